// YOLOXHeadWraper_10797547782861
// MI455X (gfx1250) — compile-verified
//
#include <hip/hip_runtime.h>
#include <hip/hip_bf16.h>
#include <math.h>

// ---------------- problem constants (match reference) ----------------
#define BATCH    8
#define NCLS     80
#define NTOT     8400        // 80*80 + 40*40 + 20*20
#define HW0      6400
#define HW1      1600
#define HW2      400
#define KEEP     100
#define SCORE_THR 0.01f
#define IOU_THR   0.65f

typedef __attribute__((ext_vector_type(2))) float v2f;
typedef __attribute__((ext_vector_type(8))) float v8f;

// ---------------------------------------------------------------------
// Kernel 1: box decode via V_WMMA_F32_16X16X4_F32.
// [x1,y1,x2,y2] = [cx,cy,w,h] * Mt  where Mt (4x4) is the center->corner
// linear map. Each wave decodes 16 candidates with one WMMA.
// A layout (16x4 f32, ISA 7.12.2): lanes 0-15 hold row M=lane, VGPR0=K0,
// VGPR1=K1; lanes 16-31 hold row M=lane-16, VGPR0=K2, VGPR1=K3.
// ---------------------------------------------------------------------
__global__ void __launch_bounds__(256)
decode_kernel(const float* __restrict__ bbox0,
              const float* __restrict__ bbox1,
              const float* __restrict__ bbox2,
              float* __restrict__ boxes /* [B*NTOT,4] xyxy */) {
  int tid  = blockIdx.x * blockDim.x + threadIdx.x;
  int wave = tid >> 5;
  int lane = threadIdx.x & 31;
  int half = lane >> 4;         // 0 -> provides cx,cy ; 1 -> provides w,h
  int m    = lane & 15;
  int g    = wave * 16 + m;     // candidate id in [0, B*NTOT)
  int b    = g / NTOT;
  int n    = g - b * NTOT;

  // level decomposition + prior (x*s, y*s, s, s)
  int hw, W, s;
  const float* bp;
  size_t base;
  if (n < HW0)            { hw = n;        W = 80; s = 8;
                            bp = bbox0; base = (size_t)b * 4 * HW0 + hw; }
  else if (n < HW0 + HW1) { hw = n - HW0;  W = 40; s = 16;
                            bp = bbox1; base = (size_t)b * 4 * HW1 + hw; }
  else                    { hw = n - HW0 - HW1; W = 20; s = 32;
                            bp = bbox2; base = (size_t)b * 4 * HW2 + hw; }
  int HWl = (n < HW0) ? HW0 : (n < HW0 + HW1 ? HW1 : HW2);
  int y = hw / W, x = hw - y * W;
  float fs = (float)s;

  float p0 = bp[base + (size_t)(half ? 2 : 0) * HWl];
  float p1 = bp[base + (size_t)(half ? 3 : 1) * HWl];

  float a0, a1;
  if (half == 0) { a0 = p0 * fs + (float)x * fs; a1 = p1 * fs + (float)y * fs; }
  else           { a0 = __expf(p0) * fs;         a1 = __expf(p1) * fs; }
  v2f A; A.x = a0; A.y = a1;

  // B = Mt zero-padded to 4x16.  Mt rows (K): cx:[1,0,1,0] cy:[0,1,0,1]
  // w:[-.5,0,.5,0] h:[0,-.5,0,.5]
  float b0 = 0.f, b1 = 0.f;
  if (m < 4) {
    const float Mt[4][4] = {{1.f, 0.f, 1.f, 0.f},
                            {0.f, 1.f, 0.f, 1.f},
                            {-0.5f, 0.f, 0.5f, 0.f},
                            {0.f, -0.5f, 0.f, 0.5f}};
    b0 = Mt[half * 2 + 0][m];
    b1 = Mt[half * 2 + 1][m];
  }
  v2f Bv; Bv.x = b0; Bv.y = b1;

  v8f C = {};
  v8f D = __builtin_amdgcn_wmma_f32_16x16x4_f32(
      /*neg_a=*/false, A, /*neg_b=*/false, Bv,
      /*c_mod=*/(short)0, C, /*reuse_a=*/false, /*reuse_b=*/false);

  // D layout: VGPR r, lanes 0-15 -> row M=r col=lane; lanes 16-31 -> M=r+8.
  if (m < 4) {
#pragma unroll
    for (int r = 0; r < 8; ++r) {
      long gr = (long)wave * 16 + r + half * 8;
      boxes[gr * 4 + m] = D[r];
    }
  }
}

// ---------------------------------------------------------------------
// Kernel 2: objectness sigmoid
// ---------------------------------------------------------------------
__global__ void obj_kernel(const float* __restrict__ obj0,
                           const float* __restrict__ obj1,
                           const float* __restrict__ obj2,
                           float* __restrict__ objsig) {
  int t = blockIdx.x * blockDim.x + threadIdx.x;
  if (t >= BATCH * NTOT) return;
  int b = t / NTOT, n = t - b * NTOT;
  float v;
  if (n < HW0)            v = obj0[(size_t)b * HW0 + n];
  else if (n < HW0 + HW1) v = obj1[(size_t)b * HW1 + (n - HW0)];
  else                    v = obj2[(size_t)b * HW2 + (n - HW0 - HW1)];
  objsig[t] = 1.f / (1.f + __expf(-v));
}

// ---------------------------------------------------------------------
// Kernel 3: per-(image,class) greedy NMS.
// Class logits are staged into LDS with CDNA5 async global->LDS DMA.
// ---------------------------------------------------------------------
__global__ void __launch_bounds__(256)
nms_kernel(const float* __restrict__ cls0,
           const float* __restrict__ cls1,
           const float* __restrict__ cls2,
           const float* __restrict__ objsig,
           const float* __restrict__ boxes,
           float* __restrict__ kept_s,   /* [B*NCLS*KEEP]   */
           float* __restrict__ kept_b) { /* [B*NCLS*KEEP*4] */
  __shared__ __align__(16) float s_scores[NTOT];
  __shared__ float red_val[256];
  __shared__ int   red_idx[256];

  const int tid = threadIdx.x;
  const int b = blockIdx.x / NCLS;
  const int c = blockIdx.x - b * NCLS;

  const float* seg0 = cls0 + ((size_t)b * NCLS + c) * HW0;
  const float* seg1 = cls1 + ((size_t)b * NCLS + c) * HW1;
  const float* seg2 = cls2 + ((size_t)b * NCLS + c) * HW2;

  // ---- async stage: 2100 x b128 (= NTOT floats) into LDS ----
  for (int t = tid; t < (NTOT / 4); t += 256) {
    const float* gp;
    int fofs;                   // float offset in s_scores
    if (t < HW0 / 4)                 { gp = seg0 + (size_t)t * 4;              fofs = t * 4; }
    else if (t < (HW0 + HW1) / 4)    { gp = seg1 + (size_t)(t - HW0/4) * 4;    fofs = t * 4; }
    else                             { gp = seg2 + (size_t)(t - (HW0+HW1)/4) * 4; fofs = t * 4; }
    unsigned           lo = (unsigned)(uintptr_t)(const void*)(s_scores + fofs);
    unsigned long long ga = (unsigned long long)(uintptr_t)gp;
    asm volatile("global_load_async_to_lds_b128 %0, %1, off"
                 :: "v"(lo), "v"(ga) : "memory");
  }
  asm volatile("s_wait_asynccnt 0" ::: "memory");
  __syncthreads();

  // ---- fused sigmoid(cls)*sigmoid(obj) + threshold, in place ----
  for (int i = tid; i < NTOT; i += 256) {
    float sc = 1.f / (1.f + __expf(-s_scores[i]));
    sc *= objsig[(size_t)b * NTOT + i];
    s_scores[i] = (sc > SCORE_THR) ? sc : -1.f;
  }
  __syncthreads();

  // ---- 100 greedy rounds: argmax + IoU suppression ----
  for (int it = 0; it < KEEP; ++it) {
    float v = -2.f; int vi = 0;
    for (int i = tid; i < NTOT; i += 256) {
      float sc = s_scores[i];
      if (sc > v || (sc == v && i < vi)) { v = sc; vi = i; }
    }
    red_val[tid] = v; red_idx[tid] = vi;
    __syncthreads();
#pragma unroll
    for (int off = 128; off > 0; off >>= 1) {
      if (tid < off) {
        float ov = red_val[tid + off]; int oi = red_idx[tid + off];
        if (ov > red_val[tid] || (ov == red_val[tid] && oi < red_idx[tid])) {
          red_val[tid] = ov; red_idx[tid] = oi;
        }
      }
      __syncthreads();
    }
    float bestv = red_val[0];
    int   besti = red_idx[0];
    size_t slot = ((size_t)b * NCLS + c) * KEEP + it;

    if (bestv > 0.f) {
      const float* bb = boxes + ((size_t)b * NTOT + besti) * 4;
      float bx1 = bb[0], by1 = bb[1], bx2 = bb[2], by2 = bb[3];
      float barea = (bx2 - bx1) * (by2 - by1);
      for (int i = tid; i < NTOT; i += 256) {
        const float* ob = boxes + ((size_t)b * NTOT + i) * 4;
        float x1 = ob[0], y1 = ob[1], x2 = ob[2], y2 = ob[3];
        float iw = fminf(bx2, x2) - fmaxf(bx1, x1);
        float ih = fminf(by2, y2) - fmaxf(by1, y1);
        iw = fmaxf(iw, 0.f); ih = fmaxf(ih, 0.f);
        float inter = iw * ih;
        float a2 = (x2 - x1) * (y2 - y1);
        float iou = inter / (barea + a2 - inter + 1e-6f);
        if (iou > IOU_THR) s_scores[i] = -1.f;
      }
      if (tid == 0) {
        kept_s[slot] = bestv;
        kept_b[slot * 4 + 0] = bx1; kept_b[slot * 4 + 1] = by1;
        kept_b[slot * 4 + 2] = bx2; kept_b[slot * 4 + 3] = by2;
        s_scores[besti] = -1.f;   // guard degenerate zero-area self-IoU
      }
    } else {
      if (tid == 0) {
        kept_s[slot] = -1.f;
        kept_b[slot * 4 + 0] = 0.f; kept_b[slot * 4 + 1] = 0.f;
        kept_b[slot * 4 + 2] = 0.f; kept_b[slot * 4 + 3] = 0.f;
      }
    }
    __syncthreads();
  }
}

// ---------------------------------------------------------------------
// Kernel 4: global keepTopK=100 per image over NCLS*KEEP candidates.
// Output layout (flat, reference tuple order):
//   [0,B)                    num_detected (as float)
//   [B, B+B*100*4)           proposals
//   [B+B*400, B+B*500)       scores
//   [B+B*500, B+B*600)       class ids (as float, -1 invalid)
// ---------------------------------------------------------------------
__global__ void __launch_bounds__(256)
final_kernel(const float* __restrict__ kept_s,
             const float* __restrict__ kept_b,
             float* __restrict__ out) {
  __shared__ float fs[NCLS * KEEP];
  __shared__ float red_val[256];
  __shared__ int   red_idx[256];

  const int tid = threadIdx.x;
  const int b = blockIdx.x;
  const int NF = NCLS * KEEP;

  for (int i = tid; i < NF; i += 256)
    fs[i] = kept_s[(size_t)b * NF + i];
  __syncthreads();

  float* out_boxes = out + BATCH + (size_t)b * KEEP * 4;
  float* out_score = out + BATCH + (size_t)BATCH * KEEP * 4 + (size_t)b * KEEP;
  float* out_cls   = out + BATCH + (size_t)BATCH * KEEP * 5 + (size_t)b * KEEP;

  int cnt = 0;
  for (int it = 0; it < KEEP; ++it) {
    float v = -3.f; int vi = 0;
    for (int i = tid; i < NF; i += 256) {
      float sc = fs[i];
      if (sc > v || (sc == v && i < vi)) { v = sc; vi = i; }
    }
    red_val[tid] = v; red_idx[tid] = vi;
    __syncthreads();
#pragma unroll
    for (int off = 128; off > 0; off >>= 1) {
      if (tid < off) {
        float ov = red_val[tid + off]; int oi = red_idx[tid + off];
        if (ov > red_val[tid] || (ov == red_val[tid] && oi < red_idx[tid])) {
          red_val[tid] = ov; red_idx[tid] = oi;
        }
      }
      __syncthreads();
    }
    float bestv = red_val[0];
    int   besti = red_idx[0];
    if (tid == 0) {
      if (bestv > 0.f) {
        const float* bb = kept_b + ((size_t)b * NF + besti) * 4;
        out_boxes[it * 4 + 0] = bb[0]; out_boxes[it * 4 + 1] = bb[1];
        out_boxes[it * 4 + 2] = bb[2]; out_boxes[it * 4 + 3] = bb[3];
        out_score[it] = bestv;
        out_cls[it]   = (float)(besti / KEEP);
        ++cnt;
      } else {
        out_boxes[it * 4 + 0] = 0.f; out_boxes[it * 4 + 1] = 0.f;
        out_boxes[it * 4 + 2] = 0.f; out_boxes[it * 4 + 3] = 0.f;
        out_score[it] = 0.f;
        out_cls[it]   = -1.f;
      }
      fs[besti] = -2.f;   // remove from further selection
    }
    __syncthreads();
  }
  if (tid == 0) out[b] = (float)cnt;
}

// ---------------------------------------------------------------------
extern "C" void kernel_launch(void* const* d_in, const int* in_sizes, int n_in,
                              void* d_out, int out_size, void* d_ws, size_t ws_size,
                              hipStream_t stream) {
  const float* cls0  = (const float*)d_in[0];
  const float* bbox0 = (const float*)d_in[1];
  const float* obj0  = (const float*)d_in[2];
  const float* cls1  = (const float*)d_in[3];
  const float* bbox1 = (const float*)d_in[4];
  const float* obj1  = (const float*)d_in[5];
  const float* cls2  = (const float*)d_in[6];
  const float* bbox2 = (const float*)d_in[7];
  const float* obj2  = (const float*)d_in[8];

  float* ws      = (float*)d_ws;
  float* boxes   = ws;                                  // B*NTOT*4   = 268800
  float* objsig  = ws + (size_t)BATCH * NTOT * 4;       // B*NTOT     =  67200
  float* kept_s  = objsig + (size_t)BATCH * NTOT;       // B*NC*K     =  64000
  float* kept_b  = kept_s + (size_t)BATCH * NCLS * KEEP;// B*NC*K*4   = 256000

  // decode: B*NTOT candidates, 16 per wave -> 4200 waves -> 525 blocks x 256
  decode_kernel<<<(BATCH * NTOT / 16 * 32) / 256, 256, 0, stream>>>(
      bbox0, bbox1, bbox2, boxes);
  obj_kernel<<<(BATCH * NTOT + 255) / 256, 256, 0, stream>>>(
      obj0, obj1, obj2, objsig);
  nms_kernel<<<BATCH * NCLS, 256, 0, stream>>>(
      cls0, cls1, cls2, objsig, boxes, kept_s, kept_b);
  final_kernel<<<BATCH, 256, 0, stream>>>(kept_s, kept_b, (float*)d_out);
}